// Preprocessor_21182778704531
// MI455X (gfx1250) — compile-verified
//
#include <hip/hip_runtime.h>
#include <math.h>

// Problem constants (from reference)
#define L_LEN 3750
#define NT    256
#define NWAVE (NT / 32)
#define CHUNK 15          // 250 threads * 15 = 3750, threads 250..255 idle
#define H1    125
#define K1    251
#define H2    250
#define K2    501
#define EPSF  1e-5f

typedef unsigned int u32x4 __attribute__((ext_vector_type(4)));
typedef int          i32x8 __attribute__((ext_vector_type(8)));
typedef int          i32x4 __attribute__((ext_vector_type(4)));

// Build the TDM descriptor (groups 0,1) for a 1-D row of L_LEN f32 elements.
__device__ __forceinline__ void tdm_row_desc(unsigned long long ga, unsigned lds,
                                             u32x4& g0, i32x8& g1) {
  g0[0] = 1u;                                         // count=1, user descriptor
  g0[1] = lds;                                        // lds_addr @ [63:32]
  g0[2] = (unsigned)(ga & 0xFFFFFFFFull);             // global_addr lo
  g0[3] = (unsigned)((ga >> 32) & 0x01FFFFFFull)      // global_addr[56:32]
        | (2u << 30);                                 // type=2 ("image")
  g1[0] = (int)(2u << 16);                            // data_size=4B, no multicast
  g1[1] = (int)(((unsigned)L_LEN & 0xFFFFu) << 16);   // tensor_dim0 lo16 @ [63:48]
  g1[2] = (int)(((unsigned)L_LEN >> 16) | (1u << 16));// tensor_dim0 hi16, tensor_dim1=1
  g1[3] = (int)((unsigned)L_LEN << 16);               // tile_dim0=3750 @ [127:112]
  g1[4] = 1;                                          // tile_dim1=1, tile_dim2=0
  g1[5] = L_LEN;                                      // tensor_dim0_stride
  g1[6] = 0;
  g1[7] = 0;
}

// Block-wide exclusive scan (one float per thread) via wave32 shuffle scan +
// one LDS combine of the NWAVE wave totals. Only 2 barriers.
// Returns exclusive prefix; *total receives the block-wide sum.
__device__ __forceinline__ float block_excl_scan(float own, float* swt, int t,
                                                 float* total) {
  __syncthreads();                 // protect swt[] against readers from prior use
  const int lane = t & 31;
  const int w = t >> 5;
  float v = own;                   // becomes inclusive prefix within wave
#pragma unroll
  for (int d = 1; d < 32; d <<= 1) {
    float n = __shfl_up(v, d, 32);
    if (lane >= d) v += n;
  }
  if (lane == 31) swt[w] = v;      // wave total
  __syncthreads();
  float woff = 0.0f, tot = 0.0f;
#pragma unroll
  for (int wi = 0; wi < NWAVE; ++wi) {
    float s = swt[wi];
    if (wi < w) woff += s;
    tot += s;
  }
  *total = tot;
  return v + woff - own;           // exclusive prefix
}

__global__ __launch_bounds__(NT) void preprocessor_kernel(
    const float* __restrict__ x, float* __restrict__ out) {
  __shared__ float sY[L_LEN];       // raw -> normalized y -> z -> out
  __shared__ float sC[L_LEN + 1];   // exclusive prefix sums (reused)
  __shared__ float sWT[NWAVE];      // wave totals for scans
  __shared__ float sRed[2 * NWAVE];
  __shared__ float sStat[2];

  const int t = threadIdx.x;
  const int row = blockIdx.x;
  const float* xrow = x + (size_t)row * L_LEN;
  float* orow = out + (size_t)row * L_LEN;

  // ---------- Phase A: TDM bulk DMA of the whole row into LDS ----------
  if (t == 0) {
    u32x4 g0; i32x8 g1;
    tdm_row_desc((unsigned long long)(uintptr_t)xrow,
                 (unsigned)(uintptr_t)(&sY[0]), g0, g1);
    i32x4 gz4 = (i32x4)0;
    i32x8 gz8 = (i32x8)0;
    __builtin_amdgcn_tensor_load_to_lds(g0, g1, gz4, gz4, gz8, 0);
    __builtin_amdgcn_s_wait_tensorcnt(0);
  }
  __syncthreads();

  const int base = t * CHUNK;
  int cnt = L_LEN - base;
  cnt = cnt < 0 ? 0 : (cnt > CHUNK ? CHUNK : cnt);

  // ---------- Phase B: mean / unbiased std ----------
  float s1 = 0.0f, s2 = 0.0f;
  for (int j = 0; j < cnt; ++j) {
    float v = sY[base + j];
    s1 += v;
    s2 += v * v;
  }
#pragma unroll
  for (int o = 16; o > 0; o >>= 1) {
    s1 += __shfl_down(s1, o, 32);
    s2 += __shfl_down(s2, o, 32);
  }
  if ((t & 31) == 0) { sRed[t >> 5] = s1; sRed[NWAVE + (t >> 5)] = s2; }
  __syncthreads();
  if (t == 0) {
    float S1 = 0.0f, S2 = 0.0f;
#pragma unroll
    for (int w = 0; w < NWAVE; ++w) { S1 += sRed[w]; S2 += sRed[NWAVE + w]; }
    float mean = S1 / (float)L_LEN;
    float var = (S2 - (float)L_LEN * mean * mean) / (float)(L_LEN - 1);
    var = var > 0.0f ? var : 0.0f;
    sStat[0] = mean;
    sStat[1] = 1.0f / (sqrtf(var) + EPSF);
  }
  __syncthreads();
  const float mean = sStat[0];
  const float inv  = sStat[1];

  // ---------- Phase C: z-normalize, exclusive prefix sum of y ----------
  float run = 0.0f;
  for (int j = 0; j < cnt; ++j) {
    float v = (sY[base + j] - mean) * inv;
    sY[base + j] = v;
    run += v;
  }
  float tot0;
  float off0 = block_excl_scan(run, sWT, t, &tot0);
  {
    float r = off0;
    for (int j = 0; j < cnt; ++j) { sC[base + j] = r; r += sY[base + j]; }
  }
  if (t == 0) sC[L_LEN] = tot0;
  __syncthreads();

  // ---------- Phase D: subtract moving average (K=251) ----------
  float run2 = 0.0f;
  for (int j = 0; j < cnt; ++j) {
    int i = base + j;
    int hi = i + (H1 + 1); hi = hi > L_LEN ? L_LEN : hi;
    int lo = i - H1;       lo = lo < 0 ? 0 : lo;
    float ma = (sC[hi] - sC[lo]) * (1.0f / (float)K1);
    float z = sY[i] - ma;
    sY[i] = z;
    run2 += z * z;
  }
  // scan of z^2 chunk totals (leading barrier inside covers all sC reads above)
  float tot2;
  float off2 = block_excl_scan(run2, sWT, t, &tot2);
  {
    float r = off2;
    for (int j = 0; j < cnt; ++j) {
      float z = sY[base + j];
      sC[base + j] = r;
      r += z * z;
    }
  }
  if (t == 0) sC[L_LEN] = tot2;
  __syncthreads();

  // ---------- Phase E: rescale by moving std (K=501) ----------
  for (int j = 0; j < cnt; ++j) {
    int i = base + j;
    int hi = i + (H2 + 1); hi = hi > L_LEN ? L_LEN : hi;
    int lo = i - H2;       lo = lo < 0 ? 0 : lo;
    float mv = (sC[hi] - sC[lo]) * (1.0f / (float)K2);
    mv = mv > 0.0f ? mv : 0.0f;
    sY[i] = sY[i] / (sqrtf(mv) + EPSF);
  }
  __syncthreads();

  // ---------- Phase F: TDM bulk DMA of the result row back to HBM ----------
  if (t == 0) {
    u32x4 g0; i32x8 g1;
    tdm_row_desc((unsigned long long)(uintptr_t)orow,
                 (unsigned)(uintptr_t)(&sY[0]), g0, g1);
    i32x4 gz4 = (i32x4)0;
    i32x8 gz8 = (i32x8)0;
    __builtin_amdgcn_tensor_store_from_lds(g0, g1, gz4, gz4, gz8, 0);
    // No explicit wait needed: S_ENDPGM performs an implicit wait-idle that
    // drains TENSORcnt before the wave (and its LDS) is released.
  }
}

extern "C" void kernel_launch(void* const* d_in, const int* in_sizes, int n_in,
                              void* d_out, int out_size, void* d_ws, size_t ws_size,
                              hipStream_t stream) {
  const float* x = (const float*)d_in[0];
  float* out = (float*)d_out;
  const int rows = in_sizes[0] / L_LEN;  // 8192
  preprocessor_kernel<<<dim3(rows), dim3(NT), 0, stream>>>(x, out);
}